// ScaledDotProductAttention_20744692040185
// MI455X (gfx1250) — compile-verified
//
#include <hip/hip_runtime.h>
#include <hip/hip_bf16.h>

// Problem constants (from reference setup_inputs)
#define BB 4
#define HH 8
#define SS 2048
#define DD 64
#define MT 16                 // query rows per workgroup
#define NWAVES 4
#define THREADS (NWAVES * 32)
#define LDS_STRIDE (SS + 4)   // pad rows by 4 floats -> conflict-free phase-3 column access

#define AS1 __attribute__((address_space(1)))
#define AS3 __attribute__((address_space(3)))

typedef __attribute__((ext_vector_type(2))) float v2f;
typedef __attribute__((ext_vector_type(4))) float v4f;
typedef __attribute__((ext_vector_type(8))) float v8f;
typedef __attribute__((ext_vector_type(4))) int   v4i;

__global__ __launch_bounds__(THREADS)
void sdpa_f32_wmma_kernel(const float* __restrict__ q,
                          const float* __restrict__ k,
                          const float* __restrict__ v,
                          const int*   __restrict__ mask,
                          const float* __restrict__ bias,
                          float* __restrict__ out,
                          float* __restrict__ attn) {
  __shared__ float sQ[MT * DD];          // 4 KB Q tile
  extern __shared__ float sP[];          // MT * LDS_STRIDE floats (~128 KB) score slab

  const int bh   = blockIdx.y;           // 0 .. B*H-1
  const int h    = bh & (HH - 1);
  const int q0   = blockIdx.x * MT;      // query row base
  const int tid  = threadIdx.x;
  const int wave = tid >> 5;
  const int lane = tid & 31;

  const float* qbase = q    + ((size_t)bh * SS + q0) * DD;
  const float* kbase = k    + (size_t)bh * SS * DD;
  const float* vbase = v    + (size_t)bh * SS * DD;
  const float* bbase = bias + (size_t)h * SS * SS + (size_t)q0 * SS;
  const int*   mbase = mask + (size_t)h * SS * SS + (size_t)q0 * SS;
  float* attn_base = attn + ((size_t)bh * SS + q0) * SS;
  float* out_base  = out  + ((size_t)bh * SS + q0) * DD;

  // ---- stage Q tile (16x64 f32 = 4 KB) into LDS via async global->LDS DMA ----
#if __has_builtin(__builtin_amdgcn_global_load_async_to_lds_b128)
  for (int i = tid; i < MT * DD / 4; i += THREADS) {
    AS1 v4i* g = (AS1 v4i*)(unsigned long long)(const void*)(qbase + 4 * i);
    AS3 v4i* l =
        (AS3 v4i*)(unsigned long long)(unsigned int)(unsigned long long)(const void*)(sQ + 4 * i);
    __builtin_amdgcn_global_load_async_to_lds_b128(g, l, 0, 0);
  }
#if __has_builtin(__builtin_amdgcn_s_wait_asynccnt)
  __builtin_amdgcn_s_wait_asynccnt(0);
#else
  asm volatile("s_wait_asynccnt 0x0" ::: "memory");
#endif
#else
  for (int i = tid; i < MT * DD; i += THREADS) sQ[i] = qbase[i];
#endif
  __syncthreads();

  // ---- build WMMA A fragments of Q: 16 chunks of K=4 over D=64 ----
  // A 16x4 f32 layout: lanes 0-15 -> M=lane, K={4c+0,4c+1}; lanes 16-31 -> M=lane-16, K={4c+2,4c+3}
  const int am = lane & 15;
  const int ar = (lane < 16) ? 0 : 2;
  v2f aq[16];
#pragma unroll
  for (int c = 0; c < 16; ++c) {
    aq[c].x = sQ[am * DD + 4 * c + ar + 0];
    aq[c].y = sQ[am * DD + 4 * c + ar + 1];
  }

  // ================= Phase 1: scores = QK^T/8 + bias, masked-fill =================
  // B 4x16 f32 layout: lane column n=lane&15; rows {ar, ar+1}
  const int bn = lane & 15;
  for (int jt = wave; jt < SS / 16; jt += NWAVES) {
    const int n0 = jt * 16;

    // prefetch next tile's bias/mask lines into cache (global_prefetch_b8)
    if (jt + NWAVES < SS / 16) {
      const int nn = (jt + NWAVES) * 16;
      __builtin_prefetch(bbase + (size_t)(lane & 15) * SS + nn, 0, 0);
      __builtin_prefetch(mbase + (size_t)(lane & 15) * SS + nn, 0, 0);
    }

    v8f acc = {};
#pragma unroll
    for (int c = 0; c < 16; ++c) {
      const float* kp = kbase + (size_t)(n0 + bn) * DD + 4 * c + ar;
      v2f bk;
      bk.x = kp[0];
      bk.y = kp[1];
      acc = __builtin_amdgcn_wmma_f32_16x16x4_f32(
          /*neg_a=*/false, aq[c], /*neg_b=*/false, bk,
          /*c_mod=*/(short)0, acc, /*reuse_a=*/false, /*reuse_b=*/false);
    }
    // C/D layout: VGPR i -> row m = i + (lane<16 ? 0 : 8), col = n0 + (lane&15)
#pragma unroll
    for (int i = 0; i < 8; ++i) {
      const int m   = i + ((lane < 16) ? 0 : 8);
      const int col = n0 + bn;
      float s = acc[i] * 0.125f + bbase[(size_t)m * SS + col];
      if (mbase[(size_t)m * SS + col] == 0) s = -__builtin_inff();
      sP[m * LDS_STRIDE + col] = s;
    }
  }
  __syncthreads();

  // ================= Phase 2: rowwise softmax (8 lanes per row, float4) ==========
  {
    const int row = tid >> 3;   // 0..15
    const int sub = tid & 7;
    float* rp = sP + row * LDS_STRIDE;

    float mx = -__builtin_inff();
    for (int g = sub; g < SS / 4; g += 8) {
      v4f x = *(const v4f*)(rp + 4 * g);
      mx = fmaxf(mx, fmaxf(fmaxf(x.x, x.y), fmaxf(x.z, x.w)));
    }
    mx = fmaxf(mx, __shfl_xor(mx, 1, 32));
    mx = fmaxf(mx, __shfl_xor(mx, 2, 32));
    mx = fmaxf(mx, __shfl_xor(mx, 4, 32));

    float sum = 0.0f;
    for (int g = sub; g < SS / 4; g += 8) {
      v4f x = *(const v4f*)(rp + 4 * g);
      x.x = __expf(x.x - mx);
      x.y = __expf(x.y - mx);
      x.z = __expf(x.z - mx);
      x.w = __expf(x.w - mx);
      *(v4f*)(rp + 4 * g) = x;
      sum += (x.x + x.y) + (x.z + x.w);
    }
    sum += __shfl_xor(sum, 1, 32);
    sum += __shfl_xor(sum, 2, 32);
    sum += __shfl_xor(sum, 4, 32);
    const float inv = 1.0f / sum;

    for (int g = sub; g < SS / 4; g += 8) {
      v4f x = *(const v4f*)(rp + 4 * g);
      x *= inv;
      *(v4f*)(rp + 4 * g) = x;
      *(v4f*)(attn_base + (size_t)row * SS + 4 * g) = x;  // b128 coalesced attn stream
    }
  }
  __syncthreads();

  // ================= Phase 3: out = P @ V (K = 2048, each wave owns 16 d-cols) ====
  {
    const int d0 = wave * 16;
    v8f oacc = {};
#pragma unroll 4
    for (int c = 0; c < SS / 4; ++c) {
      v2f ap, bv;
      ap.x = sP[am * LDS_STRIDE + 4 * c + ar + 0];
      ap.y = sP[am * LDS_STRIDE + 4 * c + ar + 1];
      const float* vp = vbase + (size_t)(4 * c + ar) * DD + d0 + bn;
      bv.x = vp[0];
      bv.y = vp[DD];
      oacc = __builtin_amdgcn_wmma_f32_16x16x4_f32(
          false, ap, false, bv, (short)0, oacc, false, false);
    }
#pragma unroll
    for (int i = 0; i < 8; ++i) {
      const int m = i + ((lane < 16) ? 0 : 8);
      out_base[(size_t)m * DD + d0 + bn] = oacc[i];
    }
  }
}

extern "C" void kernel_launch(void* const* d_in, const int* in_sizes, int n_in,
                              void* d_out, int out_size, void* d_ws, size_t ws_size,
                              hipStream_t stream) {
  (void)in_sizes; (void)n_in; (void)d_ws; (void)ws_size; (void)out_size;
  const float* q    = (const float*)d_in[0];
  const float* k    = (const float*)d_in[1];
  const float* v    = (const float*)d_in[2];
  const int*   mask = (const int*)d_in[3];
  const float* bias = (const float*)d_in[4];

  float* out  = (float*)d_out;                              // [B,H,S,D]
  float* attn = out + (size_t)BB * HH * SS * DD;            // [B,H,S,S]

  dim3 grid(SS / MT, BB * HH);
  dim3 block(THREADS);
  const size_t lds_bytes = (size_t)MT * LDS_STRIDE * sizeof(float);
  sdpa_f32_wmma_kernel<<<grid, block, lds_bytes, stream>>>(q, k, v, mask, bias, out, attn);
}